// WNO2d_64390149702242
// MI455X (gfx1250) — compile-verified
//
#include <hip/hip_runtime.h>
#include <math.h>

// ============================================================================
// WNO2d for MI455X (gfx1250). fp32 end-to-end; all matmul-like stages use
// V_WMMA_F32_16X16X4_F32 (fp32 WMMA -> no precision loss vs reference).
// Weight-tile staging in k_mix_high uses GLOBAL_LOAD_ASYNC_TO_LDS_B128
// (ASYNCcnt path) since that tile is consumed in its memory layout.
// Workspace requirement: ~397 MB.
// ============================================================================

#define DEVINL static __device__ __forceinline__

typedef float v2f __attribute__((ext_vector_type(2)));
typedef float v8f __attribute__((ext_vector_type(8)));

DEVINL v8f vzero8() { v8f z = {0.f,0.f,0.f,0.f,0.f,0.f,0.f,0.f}; return z; }

// D = A(16x4) * B(4x16) + C  (fp32 WMMA, wave32)
DEVINL v8f wmma4(v2f a, v2f b, v8f c) {
  return __builtin_amdgcn_wmma_f32_16x16x4_f32(false, a, false, b, (short)0, c,
                                               false, false);
}

DEVINL float gelu_exact(float x) {
  return 0.5f * x * (1.0f + erff(x * 0.70710678118654752440f));
}

// ---------------- wavelet filters (db6-like, L=12) --------------------------
#define D0  (-0.00107730108499558f)
#define D1  (0.004777257511010651f)
#define D2  (0.0005538422009938016f)
#define D3  (-0.031582039318031156f)
#define D4  (0.02752286553001629f)
#define D5  (0.09750160558707936f)
#define D6  (-0.12976686756709563f)
#define D7  (-0.22626469396516913f)
#define D8  (0.3152503517092432f)
#define D9  (0.7511339080215775f)
#define D10 (0.4946238903983854f)
#define D11 (0.11154074335008017f)

// analysis: KA0 = REC_LO = reverse(DEC_LO); KA1 = REC_HI = (-1)^t * DEC_LO[t]
__constant__ float KA0[12] = { D11,D10,D9,D8,D7,D6,D5,D4,D3,D2,D1,D0 };
__constant__ float KA1[12] = { D0,-D1,D2,-D3,D4,-D5,D6,-D7,D8,-D9,D10,-D11 };
// synthesis: KS0 = DEC_LO; KS1 = DEC_HI = reverse(REC_HI)
__constant__ float KS0[12] = { D0,D1,D2,D3,D4,D5,D6,D7,D8,D9,D10,D11 };
__constant__ float KS1[12] = { -D11,D10,-D9,D8,-D7,D6,-D5,D4,-D3,D2,-D1,D0 };

// ---------------- problem constants -----------------------------------------
constexpr int NB = 8, NC = 32, S = 256;
constexpr int HP = 258;                  // padded (PAD=1)
constexpr int Hd1 = 134, Hd2 = 72, Hd3 = 41;  // DWT level sizes

// ============================================================================
// fc0 lift (4->32) fused with reflect-pad(1) into NHWC h[b][y][x][c]
// ============================================================================
__global__ __launch_bounds__(256) void k_fc0pad(
    const float* __restrict__ u, const float* __restrict__ par,
    const float* __restrict__ xin, const float* __restrict__ yin,
    const float* __restrict__ w, const float* __restrict__ bvec,
    float* __restrict__ out) {
  long idx = (long)blockIdx.x * blockDim.x + threadIdx.x;
  long total = (long)NB * HP * HP * NC;
  if (idx >= total) return;
  int c = (int)(idx & 31); long p = idx >> 5;
  int x = (int)(p % HP); p /= HP;
  int y = (int)(p % HP); int b = (int)(p / HP);
  int yy = y - 1; yy = yy < 0 ? -yy : (yy > S - 1 ? 2 * (S - 1) - yy : yy);
  int xx = x - 1; xx = xx < 0 ? -xx : (xx > S - 1 ? 2 * (S - 1) - xx : xx);
  float v0 = u[((size_t)b * S + yy) * S + xx];
  float v1 = par[((size_t)b * S + yy) * S + xx];
  float v2 = xin[(size_t)b * S + yy];
  float v3 = yin[(size_t)b * S + xx];
  const float* wr = w + (size_t)c * 4;
  out[idx] = wr[0]*v0 + wr[1]*v1 + wr[2]*v2 + wr[3]*v3 + bvec[c];
}

// ============================================================================
// bilinear interpolation of 64x64 weight kernel -> out[row][col][i][o]
// (align_corners=False, clamped edges; matches reference coords())
// ============================================================================
__global__ __launch_bounds__(256) void k_interp(
    const float* __restrict__ w,   // [i][o][64][64] (one layer slice)
    float* __restrict__ out, int Ho, int Wo) {
  long idx = (long)blockIdx.x * blockDim.x + threadIdx.x;
  long total = (long)Ho * Wo * 1024;
  if (idx >= total) return;
  int o = (int)(idx & 31); int i = (int)((idx >> 5) & 31);
  long p = idx >> 10;
  int xo = (int)(p % Wo); int yo = (int)(p / Wo);
  float sy = fmaxf((yo + 0.5f) * (64.0f / Ho) - 0.5f, 0.0f);
  int y0 = (int)floorf(sy); y0 = y0 > 63 ? 63 : y0;
  int y1 = y0 + 1 > 63 ? 63 : y0 + 1;
  float ty = fminf(fmaxf(sy - (float)y0, 0.0f), 1.0f);
  float sx = fmaxf((xo + 0.5f) * (64.0f / Wo) - 0.5f, 0.0f);
  int x0 = (int)floorf(sx); x0 = x0 > 63 ? 63 : x0;
  int x1 = x0 + 1 > 63 ? 63 : x0 + 1;
  float tx = fminf(fmaxf(sx - (float)x0, 0.0f), 1.0f);
  const float* wp = w + ((size_t)i * 32 + o) * 4096;
  float v00 = wp[y0 * 64 + x0], v01 = wp[y0 * 64 + x1];
  float v10 = wp[y1 * 64 + x0], v11 = wp[y1 * 64 + x1];
  out[idx] = (v00 * (1.f - ty) + v10 * ty) * (1.f - tx) +
             (v01 * (1.f - ty) + v11 * ty) * tx;
}

// ============================================================================
// DWT analysis passes (symmetric pad L-2=10, stride 2, 12 taps)
// ============================================================================
__global__ __launch_bounds__(256) void k_dwt_row(
    const float* __restrict__ in, float* __restrict__ out_lo,
    float* __restrict__ out_hi, int H, int W, int Wo,
    size_t loBS, size_t hiBS) {
  long idx = (long)blockIdx.x * blockDim.x + threadIdx.x;
  long total = (long)NB * H * Wo * NC;
  if (idx >= total) return;
  int c = (int)(idx & 31); long p = idx >> 5;
  int xo = (int)(p % Wo); p /= Wo;
  int y = (int)(p % H);   int b = (int)(p / H);
  const float* row = in + (((size_t)b * H + y) * W) * NC + c;
  float lo = 0.f, hi = 0.f;
#pragma unroll
  for (int t = 0; t < 12; ++t) {
    int xi = 2 * xo + t - 10;
    xi = xi < 0 ? (-1 - xi) : (xi >= W ? 2 * W - 1 - xi : xi);
    float v = row[(size_t)xi * NC];
    lo += KA0[t] * v; hi += KA1[t] * v;
  }
  size_t o = (((size_t)y * Wo + xo)) * NC + c;
  out_lo[(size_t)b * loBS + o] = lo;
  out_hi[(size_t)b * hiBS + o] = hi;
}

__global__ __launch_bounds__(256) void k_dwt_col(
    const float* __restrict__ in, float* __restrict__ out_lo,
    float* __restrict__ out_hi, int H, int W, int Ho,
    size_t loBS, size_t hiBS) {
  long idx = (long)blockIdx.x * blockDim.x + threadIdx.x;
  long total = (long)NB * Ho * W * NC;
  if (idx >= total) return;
  int c = (int)(idx & 31); long p = idx >> 5;
  int x  = (int)(p % W); p /= W;
  int yo = (int)(p % Ho); int b = (int)(p / Ho);
  const float* base = in + ((size_t)b * H * W) * NC;
  float lo = 0.f, hi = 0.f;
#pragma unroll
  for (int t = 0; t < 12; ++t) {
    int yi = 2 * yo + t - 10;
    yi = yi < 0 ? (-1 - yi) : (yi >= H ? 2 * H - 1 - yi : yi);
    float v = base[((size_t)yi * W + x) * NC + c];
    lo += KA0[t] * v; hi += KA1[t] * v;
  }
  size_t o = (((size_t)yo * W + x)) * NC + c;
  out_lo[(size_t)b * loBS + o] = lo;
  out_hi[(size_t)b * hiBS + o] = hi;
}

// ============================================================================
// IDWT synthesis (zero-upsample x2, pad 1, 12 taps; out length = 2n-10)
// ============================================================================
__global__ __launch_bounds__(256) void k_idwt_col(
    const float* __restrict__ lo, const float* __restrict__ hi,
    float* __restrict__ out, int n, int W, int Ho,
    size_t loBS, size_t hiBS) {
  long idx = (long)blockIdx.x * blockDim.x + threadIdx.x;
  long total = (long)NB * Ho * W * NC;
  if (idx >= total) return;
  int c = (int)(idx & 31); long p = idx >> 5;
  int x  = (int)(p % W); p /= W;
  int yo = (int)(p % Ho); int b = (int)(p / Ho);
  const float* lb = lo + (size_t)b * loBS;
  const float* hb = hi + (size_t)b * hiBS;
  float acc = 0.f;
#pragma unroll
  for (int t = 0; t < 12; ++t) {
    int pos = yo + t - 1;
    if (pos >= 0 && !(pos & 1)) {
      int m = pos >> 1;
      if (m < n) {
        size_t o = ((size_t)m * W + x) * NC + c;
        acc += KS0[t] * lb[o] + KS1[t] * hb[o];
      }
    }
  }
  out[(((size_t)b * Ho + yo) * W + x) * NC + c] = acc;
}

__global__ __launch_bounds__(256) void k_idwt_row(
    const float* __restrict__ lo, const float* __restrict__ hi,
    float* __restrict__ out, int H, int n, int Wo,
    size_t loBS, size_t hiBS) {
  long idx = (long)blockIdx.x * blockDim.x + threadIdx.x;
  long total = (long)NB * H * Wo * NC;
  if (idx >= total) return;
  int c = (int)(idx & 31); long p = idx >> 5;
  int xo = (int)(p % Wo); p /= Wo;
  int y  = (int)(p % H);  int b = (int)(p / H);
  const float* lb = lo + (size_t)b * loBS;
  const float* hb = hi + (size_t)b * hiBS;
  float acc = 0.f;
#pragma unroll
  for (int t = 0; t < 12; ++t) {
    int pos = xo + t - 1;
    if (pos >= 0 && !(pos & 1)) {
      int m = pos >> 1;
      if (m < n) {
        size_t o = ((size_t)y * n + m) * NC + c;
        acc += KS0[t] * lb[o] + KS1[t] * hb[o];
      }
    }
  }
  out[(((size_t)b * H + y) * Wo + xo) * NC + c] = acc;
}

// ============================================================================
// Highs channel mixing: out[b,s,y,x,o] = sum_i in[b,s,y,x,i] * wt[s,y,i,o]
// GEMM per (s,y): M = NB*W rows, K=32, N=32 using fp32 WMMA 16x16x4.
// 4 KB weight tile is staged via GLOBAL_LOAD_ASYNC_TO_LDS_B128 (it is already
// in the [k][o] layout the B fragments consume -> pure byte copy, no VGPR
// round-trip; one s_wait_asynccnt before the block barrier).
// ============================================================================
__global__ __launch_bounds__(128) void k_mix_high(
    const float* __restrict__ in, const float* __restrict__ wt,
    float* __restrict__ out, int H, int W) {
  __shared__ __align__(16) float Bm[32 * 32];   // [k][o]
  int sy = blockIdx.y;
  const float* wsy = wt + (size_t)sy * 1024;
  {
    // 4096 B = 256 x b128; 128 threads x 2 chunks
    unsigned ldsbase = (unsigned)(size_t)(&Bm[0]);
    int t = threadIdx.x;
    const float* g0 = wsy + (size_t)t * 4;
    const float* g1 = wsy + (size_t)(t + 128) * 4;
    unsigned l0 = ldsbase + (unsigned)t * 16u;
    unsigned l1 = ldsbase + (unsigned)(t + 128) * 16u;
    asm volatile("global_load_async_to_lds_b128 %0, %1, off\n\t"
                 "global_load_async_to_lds_b128 %2, %3, off"
                 :: "v"(l0), "v"(g0), "v"(l1), "v"(g1)
                 : "memory");
    asm volatile("s_wait_asynccnt 0x0" ::: "memory");
  }
  __syncthreads();
  int lane = threadIdx.x & 31, wave = threadIdx.x >> 5;
  int Mtot = NB * W;
  int r0 = (blockIdx.x * 4 + wave) * 16;
  if (r0 >= Mtot) return;                       // wave-uniform exit
  int m = lane & 15, kb = (lane >> 4) << 1, half = lane >> 4;
  int s = sy / H, y = sy - s * H;
  int r = r0 + m; if (r >= Mtot) r = Mtot - 1;  // clamp, EXEC stays full
  int bb = r / W, xx = r - bb * W;
  const float* arow = in + ((((size_t)bb * 3 + s) * H + y) * W + xx) * NC;
  __builtin_prefetch(arow, 0, 3);
  v8f acc0 = vzero8(), acc1 = vzero8();
#pragma unroll
  for (int k = 0; k < 32; k += 4) {
    v2f a;  a.x  = arow[k + kb];            a.y  = arow[k + kb + 1];
    v2f b0; b0.x = Bm[(k + kb) * 32 + m];   b0.y = Bm[(k + kb + 1) * 32 + m];
    v2f b1; b1.x = Bm[(k + kb) * 32 + 16 + m];
            b1.y = Bm[(k + kb + 1) * 32 + 16 + m];
    acc0 = wmma4(a, b0, acc0);
    acc1 = wmma4(a, b1, acc1);
  }
#pragma unroll
  for (int vr = 0; vr < 8; ++vr) {
    int rr = r0 + vr + 8 * half;
    if (rr < Mtot) {
      int b2 = rr / W, x2 = rr - b2 * W;
      float* orow = out + ((((size_t)b2 * 3 + s) * H + y) * W + x2) * NC;
      orow[m]      = acc0[vr];
      orow[16 + m] = acc1[vr];
    }
  }
}

// ============================================================================
// LL channel mixing (per-pixel 32x32 weights): one wave per pixel, M=8 (batch)
// padded to 16. out[b,y,x,o] = sum_i ll[b,y,x,i] * wi[y,x,i,o]
// ============================================================================
__global__ __launch_bounds__(128) void k_mix_ll(
    const float* __restrict__ ll, const float* __restrict__ wi,
    float* __restrict__ out, int HW) {
  int wv = blockIdx.x * (blockDim.x >> 5) + (threadIdx.x >> 5);
  if (wv >= HW) return;                         // wave-uniform
  int lane = threadIdx.x & 31;
  int m = lane & 15, kb = (lane >> 4) << 1, half = lane >> 4;
  const float* wp = wi + (size_t)wv * 1024;     // [i][o]
  int bidx = m < NB ? m : NB - 1;               // clamp pad rows
  const float* arow = ll + ((size_t)bidx * HW + wv) * NC;
  v8f acc0 = vzero8(), acc1 = vzero8();
#pragma unroll
  for (int k = 0; k < 32; k += 4) {
    v2f a;  a.x  = arow[k + kb];              a.y  = arow[k + kb + 1];
    v2f b0; b0.x = wp[(k + kb) * 32 + m];     b0.y = wp[(k + kb + 1) * 32 + m];
    v2f b1; b1.x = wp[(k + kb) * 32 + 16 + m];
            b1.y = wp[(k + kb + 1) * 32 + 16 + m];
    acc0 = wmma4(a, b0, acc0);
    acc1 = wmma4(a, b1, acc1);
  }
#pragma unroll
  for (int vr = 0; vr < 8; ++vr) {
    int rr = vr + 8 * half;
    if (rr < NB) {
      float* orow = out + ((size_t)rr * HW + wv) * NC;
      orow[m]      = acc0[vr];
      orow[16 + m] = acc1[vr];
    }
  }
}

// ============================================================================
// Pointwise conv: u2[p,o] = sum_i h[p,i]*pw[o,i] + pb[o];  M = NB*HP*HP
// (staging transposes pw -> keep VALU staging; layout is conflict-light)
// ============================================================================
__global__ __launch_bounds__(128) void k_pointwise(
    const float* __restrict__ in, const float* __restrict__ pw,
    const float* __restrict__ pb, float* __restrict__ out, int Mtot) {
  __shared__ float Bm[32 * 32];                 // Bm[k][o] = pw[o*32+k]
  __shared__ float bias[32];
  for (int t = threadIdx.x; t < 1024; t += blockDim.x) {
    int o = t & 31, k = t >> 5;
    Bm[t] = pw[o * 32 + k];
  }
  if (threadIdx.x < 32) bias[threadIdx.x] = pb[threadIdx.x];
  __syncthreads();
  int lane = threadIdx.x & 31, wave = threadIdx.x >> 5;
  int r0 = (blockIdx.x * 4 + wave) * 16;
  if (r0 >= Mtot) return;
  int m = lane & 15, kb = (lane >> 4) << 1, half = lane >> 4;
  int r = r0 + m; if (r >= Mtot) r = Mtot - 1;
  const float* arow = in + (size_t)r * NC;
  __builtin_prefetch(arow, 0, 3);
  v8f acc0 = vzero8(), acc1 = vzero8();
#pragma unroll
  for (int k = 0; k < 32; k += 4) {
    v2f a;  a.x  = arow[k + kb];            a.y  = arow[k + kb + 1];
    v2f b0; b0.x = Bm[(k + kb) * 32 + m];   b0.y = Bm[(k + kb + 1) * 32 + m];
    v2f b1; b1.x = Bm[(k + kb) * 32 + 16 + m];
            b1.y = Bm[(k + kb + 1) * 32 + 16 + m];
    acc0 = wmma4(a, b0, acc0);
    acc1 = wmma4(a, b1, acc1);
  }
#pragma unroll
  for (int vr = 0; vr < 8; ++vr) {
    int rr = r0 + vr + 8 * half;
    if (rr < Mtot) {
      out[(size_t)rr * NC + m]      = acc0[vr] + bias[m];
      out[(size_t)rr * NC + 16 + m] = acc1[vr] + bias[16 + m];
    }
  }
}

// ============================================================================
// u1 + u2, optional exact GELU, in place into h
// ============================================================================
__global__ __launch_bounds__(256) void k_add_act(
    float* __restrict__ h, const float* __restrict__ u2, long total,
    int do_gelu) {
  long idx = (long)blockIdx.x * blockDim.x + threadIdx.x;
  if (idx >= total) return;
  float v = h[idx] + u2[idx];
  if (do_gelu) v = gelu_exact(v);
  h[idx] = v;
}

// ============================================================================
// Head: crop -> fc1 (32->128, WMMA) -> GELU -> fc2 (128->1) fused.
// Exactly 524288 rows = 8192 blocks * 4 waves * 16 rows (no partial tiles).
// ============================================================================
__global__ __launch_bounds__(128) void k_head(
    const float* __restrict__ h, const float* __restrict__ fc1w,
    const float* __restrict__ fc1b, const float* __restrict__ fc2w,
    const float* __restrict__ fc2b, float* __restrict__ out) {
  __shared__ float Bm[32 * 128];                // Bm[k][n] = fc1w[n*32+k]
  __shared__ float b1s[128], w2s[128];
  for (int t = threadIdx.x; t < 4096; t += blockDim.x) {
    int n = t & 127, k = t >> 7;
    Bm[t] = fc1w[n * 32 + k];
  }
  if (threadIdx.x < 128) { b1s[threadIdx.x] = fc1b[threadIdx.x];
                           w2s[threadIdx.x] = fc2w[threadIdx.x]; }
  __syncthreads();
  int lane = threadIdx.x & 31, wave = threadIdx.x >> 5;
  int r0 = (blockIdx.x * 4 + wave) * 16;
  int m = lane & 15, kb = (lane >> 4) << 1, half = lane >> 4;
  int r = r0 + m;
  int bb = r >> 16, rem = r & 65535, yy = rem >> 8, xx = rem & 255;
  const float* arow = h + (((size_t)bb * HP + yy + 1) * HP + (xx + 1)) * NC;
  __builtin_prefetch(arow, 0, 3);
  float a0[8], a1[8];
#pragma unroll
  for (int ks = 0; ks < 8; ++ks) { a0[ks] = arow[ks * 4 + kb];
                                   a1[ks] = arow[ks * 4 + kb + 1]; }
  float partial[8] = {0.f,0.f,0.f,0.f,0.f,0.f,0.f,0.f};
#pragma unroll
  for (int t = 0; t < 8; ++t) {                 // 8 N-tiles of fc1 (128 cols)
    v8f acc = vzero8();
#pragma unroll
    for (int ks = 0; ks < 8; ++ks) {
      v2f a; a.x = a0[ks]; a.y = a1[ks];
      v2f b; b.x = Bm[(ks * 4 + kb) * 128 + t * 16 + m];
             b.y = Bm[(ks * 4 + kb + 1) * 128 + t * 16 + m];
      acc = wmma4(a, b, acc);
    }
    int n = t * 16 + m;
    float bias = b1s[n], w2 = w2s[n];
#pragma unroll
    for (int vr = 0; vr < 8; ++vr)
      partial[vr] += gelu_exact(acc[vr] + bias) * w2;
  }
  // cross-lane fc2 reduction: masks 1..8 stay within each 16-lane half
#pragma unroll
  for (int vr = 0; vr < 8; ++vr) {
    float p = partial[vr];
    p += __shfl_xor(p, 1, 32);
    p += __shfl_xor(p, 2, 32);
    p += __shfl_xor(p, 4, 32);
    p += __shfl_xor(p, 8, 32);
    partial[vr] = p;
  }
  if (m == 0) {                                 // lanes 0 (rows 0-7) & 16 (8-15)
    float b2 = fc2b[0];
#pragma unroll
    for (int vr = 0; vr < 8; ++vr)
      out[(size_t)(r0 + vr + 8 * half)] = partial[vr] + b2;
  }
}

// ============================================================================
// Host orchestration
// ============================================================================
static inline unsigned ceil_div_u(long a, long b) { return (unsigned)((a + b - 1) / b); }

extern "C" void kernel_launch(void* const* d_in, const int* in_sizes, int n_in,
                              void* d_out, int out_size, void* d_ws, size_t ws_size,
                              hipStream_t stream) {
  const float* u    = (const float*)d_in[0];
  const float* par  = (const float*)d_in[1];
  const float* xin  = (const float*)d_in[2];
  const float* yin  = (const float*)d_in[3];
  const float* fc0w = (const float*)d_in[4];
  const float* fc0b = (const float*)d_in[5];
  const float* wcw  = (const float*)d_in[6];   // (4,32,32,64,64)
  const float* pww  = (const float*)d_in[7];   // (4,32,32)
  const float* pwb  = (const float*)d_in[8];   // (4,32)
  const float* fc1w = (const float*)d_in[9];   // (128,32)
  const float* fc1b = (const float*)d_in[10];
  const float* fc2w = (const float*)d_in[11];  // (1,128)
  const float* fc2b = (const float*)d_in[12];
  float* outp = (float*)d_out;

  // ---- workspace layout (floats); total ~99.17M floats (~397 MB) ----
  float* W = (float*)d_ws;
  size_t off = 0;
  auto alloc = [&](size_t n) { float* p = W + off; off += n; return p; };
  const size_t SZ_H   = (size_t)NB * HP * HP * NC;        // 17,040,384
  const size_t SZ_T   = (size_t)NB * HP * Hd1 * NC;       //  8,850,432
  const size_t LL1    = (size_t)Hd1 * Hd1 * NC;           // per-b ll1 stride
  const size_t LL2    = (size_t)Hd2 * Hd2 * NC;
  const size_t LL3    = (size_t)Hd3 * Hd3 * NC;
  float* hbuf = alloc(SZ_H);
  float* u2   = alloc(SZ_H);
  float* tmpA = alloc(SZ_T);
  float* tmpB = alloc(SZ_T);
  float* ll1  = alloc(NB * LL1);
  float* hi1  = alloc(NB * 3 * LL1);
  float* ll2  = alloc(NB * LL2);
  float* hi2  = alloc(NB * 3 * LL2);
  float* ll3  = alloc(NB * LL3);
  float* hi3  = alloc(NB * 3 * LL3);
  float* mll3 = alloc(NB * LL3);
  float* mhi1 = alloc(NB * 3 * LL1);
  float* mhi2 = alloc(NB * 3 * LL2);
  float* mhi3 = alloc(NB * 3 * LL3);
  float* wiLL = alloc((size_t)Hd3 * Hd3 * 1024);
  float* wiH1 = alloc((size_t)3 * Hd1 * 1024);
  float* wiH2 = alloc((size_t)3 * Hd2 * 1024);
  float* wiH3 = alloc((size_t)3 * Hd3 * 1024);
  (void)ws_size; (void)in_sizes; (void)n_in; (void)out_size;

  // ---- fc0 + reflect pad ----
  {
    long total = (long)SZ_H;
    k_fc0pad<<<ceil_div_u(total, 256), 256, 0, stream>>>(u, par, xin, yin,
                                                         fc0w, fc0b, hbuf);
  }

  const long MPW = (long)NB * HP * HP;   // pointwise rows: 532,512
  const long totalH = (long)SZ_H;

  for (int lay = 0; lay < 4; ++lay) {
    const float* wc = wcw + (size_t)lay * 32 * 32 * 64 * 64;
    // interpolated mixing weights
    k_interp<<<ceil_div_u((long)Hd3 * Hd3 * 1024, 256), 256, 0, stream>>>(wc, wiLL, Hd3, Hd3);
    k_interp<<<ceil_div_u((long)3 * Hd1 * 1024, 256), 256, 0, stream>>>(wc, wiH1, 3, Hd1);
    k_interp<<<ceil_div_u((long)3 * Hd2 * 1024, 256), 256, 0, stream>>>(wc, wiH2, 3, Hd2);
    k_interp<<<ceil_div_u((long)3 * Hd3 * 1024, 256), 256, 0, stream>>>(wc, wiH3, 3, Hd3);
    // pointwise branch (reads hbuf before IDWT overwrites it)
    k_pointwise<<<ceil_div_u(MPW, 64), 128, 0, stream>>>(
        hbuf, pww + (size_t)lay * 1024, pwb + (size_t)lay * 32, u2, (int)MPW);
    // ---- DWT level 1: 258 -> 134 ----
    k_dwt_row<<<ceil_div_u((long)NB * HP * Hd1 * NC, 256), 256, 0, stream>>>(
        hbuf, tmpA, tmpB, HP, HP, Hd1, (size_t)HP * Hd1 * NC, (size_t)HP * Hd1 * NC);
    k_dwt_col<<<ceil_div_u((long)NB * Hd1 * Hd1 * NC, 256), 256, 0, stream>>>(
        tmpA, ll1, hi1 + 0 * LL1, HP, Hd1, Hd1, LL1, 3 * LL1);
    k_dwt_col<<<ceil_div_u((long)NB * Hd1 * Hd1 * NC, 256), 256, 0, stream>>>(
        tmpB, hi1 + 1 * LL1, hi1 + 2 * LL1, HP, Hd1, Hd1, 3 * LL1, 3 * LL1);
    // ---- DWT level 2: 134 -> 72 ----
    k_dwt_row<<<ceil_div_u((long)NB * Hd1 * Hd2 * NC, 256), 256, 0, stream>>>(
        ll1, tmpA, tmpB, Hd1, Hd1, Hd2, (size_t)Hd1 * Hd2 * NC, (size_t)Hd1 * Hd2 * NC);
    k_dwt_col<<<ceil_div_u((long)NB * Hd2 * Hd2 * NC, 256), 256, 0, stream>>>(
        tmpA, ll2, hi2 + 0 * LL2, Hd1, Hd2, Hd2, LL2, 3 * LL2);
    k_dwt_col<<<ceil_div_u((long)NB * Hd2 * Hd2 * NC, 256), 256, 0, stream>>>(
        tmpB, hi2 + 1 * LL2, hi2 + 2 * LL2, Hd1, Hd2, Hd2, 3 * LL2, 3 * LL2);
    // ---- DWT level 3: 72 -> 41 ----
    k_dwt_row<<<ceil_div_u((long)NB * Hd2 * Hd3 * NC, 256), 256, 0, stream>>>(
        ll2, tmpA, tmpB, Hd2, Hd2, Hd3, (size_t)Hd2 * Hd3 * NC, (size_t)Hd2 * Hd3 * NC);
    k_dwt_col<<<ceil_div_u((long)NB * Hd3 * Hd3 * NC, 256), 256, 0, stream>>>(
        tmpA, ll3, hi3 + 0 * LL3, Hd2, Hd3, Hd3, LL3, 3 * LL3);
    k_dwt_col<<<ceil_div_u((long)NB * Hd3 * Hd3 * NC, 256), 256, 0, stream>>>(
        tmpB, hi3 + 1 * LL3, hi3 + 2 * LL3, Hd2, Hd3, Hd3, 3 * LL3, 3 * LL3);
    // ---- channel mixing (WMMA) ----
    {
      int HW3 = Hd3 * Hd3;
      k_mix_ll<<<ceil_div_u(HW3, 4), 128, 0, stream>>>(ll3, wiLL, mll3, HW3);
      dim3 g1(ceil_div_u((long)NB * Hd1, 64), 3 * Hd1);
      k_mix_high<<<g1, 128, 0, stream>>>(hi1, wiH1, mhi1, Hd1, Hd1);
      dim3 g2(ceil_div_u((long)NB * Hd2, 64), 3 * Hd2);
      k_mix_high<<<g2, 128, 0, stream>>>(hi2, wiH2, mhi2, Hd2, Hd2);
      dim3 g3(ceil_div_u((long)NB * Hd3, 64), 3 * Hd3);
      k_mix_high<<<g3, 128, 0, stream>>>(hi3, wiH3, mhi3, Hd3, Hd3);
    }
    // ---- IDWT level 3: 41 -> 72 (rec -> ll2 buffer) ----
    k_idwt_col<<<ceil_div_u((long)NB * Hd2 * Hd3 * NC, 256), 256, 0, stream>>>(
        mll3, mhi3 + 0 * LL3, tmpA, Hd3, Hd3, Hd2, LL3, 3 * LL3);
    k_idwt_col<<<ceil_div_u((long)NB * Hd2 * Hd3 * NC, 256), 256, 0, stream>>>(
        mhi3 + 1 * LL3, mhi3 + 2 * LL3, tmpB, Hd3, Hd3, Hd2, 3 * LL3, 3 * LL3);
    k_idwt_row<<<ceil_div_u((long)NB * Hd2 * Hd2 * NC, 256), 256, 0, stream>>>(
        tmpA, tmpB, ll2, Hd2, Hd3, Hd2, (size_t)Hd2 * Hd3 * NC, (size_t)Hd2 * Hd3 * NC);
    // ---- IDWT level 2: 72 -> 134 (rec -> ll1 buffer) ----
    k_idwt_col<<<ceil_div_u((long)NB * Hd1 * Hd2 * NC, 256), 256, 0, stream>>>(
        ll2, mhi2 + 0 * LL2, tmpA, Hd2, Hd2, Hd1, LL2, 3 * LL2);
    k_idwt_col<<<ceil_div_u((long)NB * Hd1 * Hd2 * NC, 256), 256, 0, stream>>>(
        mhi2 + 1 * LL2, mhi2 + 2 * LL2, tmpB, Hd2, Hd2, Hd1, 3 * LL2, 3 * LL2);
    k_idwt_row<<<ceil_div_u((long)NB * Hd1 * Hd1 * NC, 256), 256, 0, stream>>>(
        tmpA, tmpB, ll1, Hd1, Hd2, Hd1, (size_t)Hd1 * Hd2 * NC, (size_t)Hd1 * Hd2 * NC);
    // ---- IDWT level 1: 134 -> 258 (u1 -> hbuf) ----
    k_idwt_col<<<ceil_div_u((long)NB * HP * Hd1 * NC, 256), 256, 0, stream>>>(
        ll1, mhi1 + 0 * LL1, tmpA, Hd1, Hd1, HP, LL1, 3 * LL1);
    k_idwt_col<<<ceil_div_u((long)NB * HP * Hd1 * NC, 256), 256, 0, stream>>>(
        mhi1 + 1 * LL1, mhi1 + 2 * LL1, tmpB, Hd1, Hd1, HP, 3 * LL1, 3 * LL1);
    k_idwt_row<<<ceil_div_u((long)NB * HP * HP * NC, 256), 256, 0, stream>>>(
        tmpA, tmpB, hbuf, HP, Hd1, HP, (size_t)HP * Hd1 * NC, (size_t)HP * Hd1 * NC);
    // ---- combine + activation ----
    k_add_act<<<ceil_div_u(totalH, 256), 256, 0, stream>>>(hbuf, u2, totalH,
                                                           lay < 3 ? 1 : 0);
  }

  // ---- head: crop + fc1 + gelu + fc2 (524288 rows exactly) ----
  k_head<<<8192, 128, 0, stream>>>(hbuf, fc1w, fc1b, fc2w, fc2b, outp);
}